// Conv2D_29566554865702
// MI455X (gfx1250) — compile-verified
//
#include <hip/hip_runtime.h>
#include <cstdint>

// ---------------------------------------------------------------------------
// 3x3 VALID conv, fp32, 8192x8192 -> 8190x8190, + bias.  Bandwidth-bound
// (AI ~2.25 FLOP/B; floor ~23us @ 23.3 TB/s).  CDNA5 path: Tensor Data Mover
// (tensor_load_to_lds, TENSORcnt) stages a 66x132 f32 tile per block with two
// descriptors (one per half-tile); compute is a rolling-window f32 FMA.
// ---------------------------------------------------------------------------

typedef unsigned int v4u __attribute__((ext_vector_type(4)));
typedef int          v8i __attribute__((ext_vector_type(8)));

static constexpr int N      = 8192;   // input edge
static constexpr int OUT_N  = 8190;   // output edge
static constexpr int TW     = 128;    // output tile width
static constexpr int THR    = 64;     // output tile height
static constexpr int LDS_W  = 132;    // floats per LDS row (= TDM tile_dim0)
static constexpr int LDS_R  = 66;     // input rows per tile (2 x 33-row DMAs)
static constexpr int HALF_R = 33;     // rows per TDM descriptor

__global__ __launch_bounds__(256) void conv3x3_tdm_kernel(
    const float* __restrict__ X,
    const float* __restrict__ Wt,
    const float* __restrict__ Bs,
    float* __restrict__ out)
{
    __shared__ __align__(16) float tile[LDS_R * LDS_W];

    const int tid = threadIdx.x;
    const int c0  = blockIdx.x * TW;
    const int r0  = blockIdx.y * THR;

    // Clamp the 132x66 input window fully inside the tensor; remember shifts.
    const int cbase  = (c0 > N - LDS_W) ? (N - LDS_W) : c0;   // <= 8060
    const int rbase  = (r0 > N - LDS_R) ? (N - LDS_R) : r0;   // <= 8126
    const int sh_c   = c0 - cbase;                            // 0 or 4
    const int sh_r   = r0 - rbase;                            // 0 or 2

    // ---- TDM: waves 0 and 1 each DMA one 33x132 half-tile into LDS --------
    if (tid < 64) {
        const int wv = __builtin_amdgcn_readfirstlane(tid >> 5);   // 0 or 1

        const unsigned lds_addr = __builtin_amdgcn_readfirstlane(
            (unsigned)(uintptr_t)&tile[0] + (unsigned)(wv * HALF_R * LDS_W * 4));

        const uint64_t ga = (uint64_t)(uintptr_t)X +
            ((uint64_t)(rbase + wv * HALF_R) * (uint64_t)N + (uint64_t)cbase) * 4ull;
        const unsigned ga_lo = __builtin_amdgcn_readfirstlane((unsigned)ga);
        const unsigned ga_hi = __builtin_amdgcn_readfirstlane((unsigned)(ga >> 32));

        // D# group 0: count=1 | lds_addr | global_addr | type=2
        v4u g0;
        g0.x = 1u;
        g0.y = lds_addr;
        g0.z = ga_lo;
        g0.w = (ga_hi & 0x01FFFFFFu) | 0x80000000u;   // addr[56:32], type=2<<30

        // D# group 1: data_size=4B, tensor 8192x8192, tile 132x33, stride 8192
        v8i g1;
        g1[0] = 0x20000;                 // data_size=2 (4 bytes)
        g1[1] = (int)(8192u << 16);      // tensor_dim0[15:0]  in [31:16]
        g1[2] = (int)(8192u << 16);      // dim0 hi=0 | tensor_dim1[15:0]
        g1[3] = (int)(132u  << 16);      // dim1 hi=0 | tile_dim0=132
        g1[4] = HALF_R;                  // tile_dim1=33, tile_dim2=0
        g1[5] = 8192;                    // tensor_dim0_stride[31:0]
        g1[6] = 0;                       // stride hi | tensor_dim1_stride lo
        g1[7] = 0;

        asm volatile("tensor_load_to_lds %0, %1"
                     :: "s"(g0), "s"(g1)
                     : "memory");
        __builtin_amdgcn_s_wait_tensorcnt(0);   // this wave's DMA complete
    }
    __syncthreads();                            // LDS tile visible to all

    // ---- Uniform weights / bias --------------------------------------------
    const float w00 = Wt[0], w01 = Wt[1], w02 = Wt[2];
    const float w10 = Wt[3], w11 = Wt[4], w12 = Wt[5];
    const float w20 = Wt[6], w21 = Wt[7], w22 = Wt[8];
    const float bias = Bs[0];

    // ---- Rolling vertical window: 1 column x 32 output rows per thread -----
    const int cx   = tid & (TW - 1);
    const int rb   = (tid >> 7) * 32;           // 0 or 32
    const int ocol = c0 + cx;
    const bool colok = (ocol < OUT_N);
    const int lc   = cx + sh_c;                 // LDS column of leftmost tap

    float t0 = 0.f, t1 = 0.f;
#pragma unroll
    for (int r = 0; r < 34; ++r) {
        const float* p = &tile[(rb + r + sh_r) * LDS_W + lc];
        const float x0 = p[0], x1 = p[1], x2 = p[2];
        const float p0 = __builtin_fmaf(w02, x2, __builtin_fmaf(w01, x1, w00 * x0));
        const float p1 = __builtin_fmaf(w12, x2, __builtin_fmaf(w11, x1, w10 * x0));
        const float p2 = __builtin_fmaf(w22, x2, __builtin_fmaf(w21, x1, w20 * x0));
        if (r >= 2) {
            const int orow = r0 + rb + r - 2;
            if (colok && orow < OUT_N) {
                const float v = t0 + p2 + bias;   // p0(r-2)+p1(r-1)+p2(r)
                __builtin_nontemporal_store(v, &out[(uint64_t)orow * OUT_N + ocol]);
            }
        }
        t0 = t1 + p1;
        t1 = p0;
    }
}

extern "C" void kernel_launch(void* const* d_in, const int* in_sizes, int n_in,
                              void* d_out, int out_size, void* d_ws, size_t ws_size,
                              hipStream_t stream) {
    (void)in_sizes; (void)n_in; (void)out_size; (void)d_ws; (void)ws_size;
    const float* X  = (const float*)d_in[0];
    const float* Wt = (const float*)d_in[1];
    const float* Bs = (const float*)d_in[2];
    float* out = (float*)d_out;

    dim3 grid((OUT_N + TW - 1) / TW,     // 64
              (OUT_N + THR - 1) / THR);  // 128
    conv3x3_tdm_kernel<<<grid, 256, 0, stream>>>(X, Wt, Bs, out);
}